// EdgeModel_2199023255661
// MI455X (gfx1250) — compile-verified
//
#include <hip/hip_runtime.h>

// EdgeModel MLP: out = relu(relu([src|tgt|eat] @ W1 + b1) @ W2 + b2)
// E = 1.6M edges, IN_D=80, HID_D=80, OUT_D=16, all fp32.
// Strategy: one 16-edge tile per wave32, fp32 WMMA 16x16x4 for both GEMMs.

#define NODE_D   32
#define EDGE_D   16
#define IN_DIM   80
#define HID_DIM  80
#define OUT_DIM  16

#define TILE_M        16
#define K_CHUNKS      20   // 80 / 4
#define N_TILES_L1    5    // 80 / 16
#define WAVES_PER_BLK 4
#define BLOCK_THREADS (WAVES_PER_BLK * 32)
#define H_STRIDE      84   // padded row stride (dwords) for H scratch

typedef __attribute__((ext_vector_type(2))) float v2f;
typedef __attribute__((ext_vector_type(8))) float v8f;

__launch_bounds__(BLOCK_THREADS)
__global__ void edge_mlp_wmma(const float* __restrict__ src,
                              const float* __restrict__ tgt,
                              const float* __restrict__ eat,
                              const float* __restrict__ w1,
                              const float* __restrict__ b1,
                              const float* __restrict__ w2,
                              const float* __restrict__ b2,
                              float* __restrict__ out,
                              int n_edges, int n_tiles)
{
    // W1 pre-swizzled into per-(ntile, kchunk) B-fragments: one b64 per lane.
    __shared__ v2f   lds_w1[N_TILES_L1 * K_CHUNKS * 32];            // 25.6 KB
    // Per-wave H transpose scratch (16 rows x 80 cols, padded).
    __shared__ float lds_h[WAVES_PER_BLK][TILE_M * H_STRIDE];       // 4 * 5376 B

    const int tid  = threadIdx.x;
    const int lane = tid & 31;
    const int wave = tid >> 5;
    const int lh   = lane >> 4;   // lane half: selects K pair / M+8
    const int ln   = lane & 15;   // N column (B/C/D) or M row (A)

    // ---- Block init: swizzle W1 (row-major [80][80]) into B-frag layout ----
    // B-frag for chunk c, tile t: lane l holds {w1[4c+2h][16t+n], w1[4c+1+2h][16t+n]}
    for (int idx = tid; idx < N_TILES_L1 * K_CHUNKS * 32; idx += BLOCK_THREADS) {
        int t  = idx / (K_CHUNKS * 32);
        int r  = idx - t * (K_CHUNKS * 32);
        int c  = r >> 5;
        int l  = r & 31;
        int k  = 4 * c + 2 * (l >> 4);
        int col = 16 * t + (l & 15);
        v2f w;
        w.x = w1[k * HID_DIM + col];
        w.y = w1[(k + 1) * HID_DIM + col];
        lds_w1[idx] = w;
    }
    __syncthreads();

    // ---- Per-wave loop invariants ----
    // W2 B-fragments (K=80 x N=16) kept in registers.
    v2f w2f[K_CHUNKS];
#pragma unroll
    for (int c = 0; c < K_CHUNKS; ++c) {
        int k = 4 * c + 2 * lh;
        v2f w;
        w.x = w2[k * OUT_DIM + ln];
        w.y = w2[(k + 1) * OUT_DIM + ln];
        w2f[c] = w;
    }
    float b1f[N_TILES_L1];
#pragma unroll
    for (int t = 0; t < N_TILES_L1; ++t) b1f[t] = b1[16 * t + ln];
    const float b2f = b2[ln];

    float* hl = &lds_h[wave][0];

    const int wave_id = blockIdx.x * WAVES_PER_BLK + wave;
    const int n_waves = gridDim.x * WAVES_PER_BLK;

    for (int tile = wave_id; tile < n_tiles; tile += n_waves) {
        const int edge0 = tile * TILE_M;
        int row = edge0 + ln;
        if (row > n_edges - 1) row = n_edges - 1;   // clamp (keeps EXEC all-1s)

        // ---- A-fragments of X = [src | tgt | eat], 16x80 fp32 ----
        // chunk c, lane l: {X[row][4c+2h], X[row][4c+2h+1]}
        v2f xa[K_CHUNKS];
#pragma unroll
        for (int c = 0; c < K_CHUNKS; ++c) {
            int g = 4 * c + 2 * lh;
            const float* p;
            int D, off;
            if (c < 8)       { p = src; D = NODE_D; off = g; }
            else if (c < 16) { p = tgt; D = NODE_D; off = g - 32; }
            else             { p = eat; D = EDGE_D; off = g - 64; }
            xa[c] = *(const v2f*)(p + (size_t)row * D + off);
        }

        // ---- Layer 1: H = relu(X @ W1 + b1), one 16-col tile at a time ----
#pragma unroll
        for (int t = 0; t < N_TILES_L1; ++t) {
            v8f acc;
#pragma unroll
            for (int i = 0; i < 8; ++i) acc[i] = b1f[t];
#pragma unroll
            for (int c = 0; c < K_CHUNKS; ++c) {
                v2f bfrag = lds_w1[(t * K_CHUNKS + c) * 32 + lane];
                acc = __builtin_amdgcn_wmma_f32_16x16x4_f32(
                        false, xa[c], false, bfrag, (short)0, acc, false, false);
            }
            // ReLU + store C-layout tile into LDS (row v+8h, col 16t+n)
#pragma unroll
            for (int v = 0; v < 8; ++v) {
                float hv = acc[v] > 0.0f ? acc[v] : 0.0f;
                hl[(v + 8 * lh) * H_STRIDE + 16 * t + ln] = hv;
            }
        }

        // ---- Layer 2: OUT = relu(H @ W2 + b2) ----
        v8f oacc;
#pragma unroll
        for (int i = 0; i < 8; ++i) oacc[i] = b2f;
#pragma unroll
        for (int c = 0; c < K_CHUNKS; ++c) {
            // A-frag from transposed H: lane l reads {H[n][4c+2h], H[n][4c+2h+1]}
            v2f afrag = *(const v2f*)(hl + ln * H_STRIDE + 4 * c + 2 * lh);
            oacc = __builtin_amdgcn_wmma_f32_16x16x4_f32(
                     false, afrag, false, w2f[c], (short)0, oacc, false, false);
        }
        // ReLU + guarded store: C-layout VGPR v -> edge (edge0+v+8h), odim n
#pragma unroll
        for (int v = 0; v < 8; ++v) {
            int e = edge0 + v + 8 * lh;
            if (e < n_edges) {
                float ov = oacc[v] > 0.0f ? oacc[v] : 0.0f;
                out[(size_t)e * OUT_DIM + ln] = ov;
            }
        }
    }
}

extern "C" void kernel_launch(void* const* d_in, const int* in_sizes, int n_in,
                              void* d_out, int out_size, void* d_ws, size_t ws_size,
                              hipStream_t stream) {
    const float* src = (const float*)d_in[0];
    const float* tgt = (const float*)d_in[1];
    const float* eat = (const float*)d_in[2];
    const float* w1  = (const float*)d_in[3];
    const float* b1  = (const float*)d_in[4];
    const float* w2  = (const float*)d_in[5];
    const float* b2  = (const float*)d_in[6];
    float* out = (float*)d_out;

    const int n_edges = in_sizes[0] / NODE_D;
    const int n_tiles = (n_edges + TILE_M - 1) / TILE_M;

    int blocks = (n_tiles + WAVES_PER_BLK - 1) / WAVES_PER_BLK;
    if (blocks > 2048) blocks = 2048;
    if (blocks < 1) blocks = 1;

    edge_mlp_wmma<<<blocks, BLOCK_THREADS, 0, stream>>>(
        src, tgt, eat, w1, b1, w2, b2, out, n_edges, n_tiles);
}